// MatchingCriterion_31834297598176
// MI455X (gfx1250) — compile-verified
//
#include <hip/hip_runtime.h>
#include <math.h>

// Problem constants (from the reference)
#define BSZ 4
#define LSZ 4096              // L (power of two: row = idx>>12)
#define NSZ 4096              // number of predicted keypoints
#define MSZ (BSZ * LSZ)       // number of gt rows = 16384

typedef float v2f __attribute__((ext_vector_type(2)));
typedef float v8f __attribute__((ext_vector_type(8)));

// ---------------------------------------------------------------------------
// K0: reset workspace (must run every call: harness replays without re-init)
// ---------------------------------------------------------------------------
__global__ void mc_init_kernel(double* acc_d, unsigned int* acc_u, int* jmin) {
    int idx = blockIdx.x * blockDim.x + threadIdx.x;
    if (idx < MSZ) jmin[idx] = LSZ;          // sentinel: "no gt>0 in this row"
    if (idx == 0) {
        acc_d[0] = 0.0;                       // coarse sum
        acc_d[1] = 0.0;                       // sum(per_norm * sel)
        acc_u[0] = 0u;                        // sel count
        acc_u[1] = 0u;                        // any_valid flag
    }
}

// ---------------------------------------------------------------------------
// K1: fused coarse CE reduction + per-row first-match (jmin) discovery.
// Streams 536 MB (the whole cost of this problem). float4 -> GLOBAL_LOAD_B128.
// Wave reduction of partials goes through V_WMMA_F32_16X16X4_F32:
//   A = {partial, 0} per lane, B = ones  =>  D[m,n] = p_m + p_{m+16}
//   lane<16 sums d[0..7] = s_0..s_7 ; lane>=16 sums s_8..s_15 ; shfl_xor(16).
// ---------------------------------------------------------------------------
__global__ void mc_coarse_kernel(const float4* __restrict__ cm,
                                 const float4* __restrict__ gt,
                                 double* __restrict__ coarse_sum,
                                 int* __restrict__ jmin) {
    const long long total4 = (long long)BSZ * LSZ * (LSZ / 4);
    const long long stride = (long long)gridDim.x * blockDim.x;
    float partial = 0.0f;

    for (long long i = (long long)blockIdx.x * blockDim.x + threadIdx.x;
         i < total4; i += stride) {
        __builtin_prefetch(&cm[i + stride], 0, 0);   // global_prefetch_b8
        __builtin_prefetch(&gt[i + stride], 0, 0);
        float4 g = gt[i];
        float4 c = cm[i];
        int row = (int)(i >> 10);                    // (4i)/4096
        int jb  = (int)((i << 2) & (LSZ - 1));       // (4i)%4096
        if (g.x != 0.0f) { partial -= g.x * logf(c.x + 1e-6f); if (g.x > 0.0f) atomicMin(&jmin[row], jb + 0); }
        if (g.y != 0.0f) { partial -= g.y * logf(c.y + 1e-6f); if (g.y > 0.0f) atomicMin(&jmin[row], jb + 1); }
        if (g.z != 0.0f) { partial -= g.z * logf(c.z + 1e-6f); if (g.z > 0.0f) atomicMin(&jmin[row], jb + 2); }
        if (g.w != 0.0f) { partial -= g.w * logf(c.w + 1e-6f); if (g.w > 0.0f) atomicMin(&jmin[row], jb + 3); }
    }

    // ---- wave32 reduction via WMMA (EXEC all-1s here: full block, converged)
    v2f a;  a[0] = partial; a[1] = 0.0f;   // A[m,0]=p_m, A[m,2]=p_{m+16}, rest 0
    v2f b1; b1[0] = 1.0f;  b1[1] = 1.0f;   // all-ones B: layout-independent
    v8f cz = {0.f,0.f,0.f,0.f,0.f,0.f,0.f,0.f};
    v8f d = __builtin_amdgcn_wmma_f32_16x16x4_f32(
        /*neg_a=*/false, a, /*neg_b=*/false, b1,
        /*c_mod=*/(short)0, cz, /*reuse_a=*/false, /*reuse_b=*/false);
    float t = d[0] + d[1] + d[2] + d[3] + d[4] + d[5] + d[6] + d[7];
    t += __shfl_xor(t, 16, 32);            // combine the two half-wave sums

    __shared__ float wsum[8];
    int lane = threadIdx.x & 31;
    int wid  = threadIdx.x >> 5;
    if (lane == 0) wsum[wid] = t;
    __syncthreads();
    if (threadIdx.x == 0) {
        float s = 0.0f;
        int nw = blockDim.x >> 5;
        for (int w = 0; w < nw; ++w) s += wsum[w];
        atomicAdd(coarse_sum, (double)s);
    }
}

// ---------------------------------------------------------------------------
// K2: fine loss. One thread per predicted keypoint n; gt0 rows tiled via LDS.
// gt0[m] = ( float(m/L), samples0[2m], samples0[2m+1] ), valid = jmin[m] < L.
// First exact match (min m) -> gather samples1[b, jmin[m]] -> norm -> gate.
// ---------------------------------------------------------------------------
__global__ void mc_fine_kernel(const float* __restrict__ samples0,
                               const float* __restrict__ samples1,
                               const float* __restrict__ mkpts0,
                               const float* __restrict__ mkpts1,
                               const int*  __restrict__ jmin,
                               double* __restrict__ sel_sum,
                               unsigned int* __restrict__ sel_cnt,
                               unsigned int* __restrict__ any_valid) {
    __shared__ float4 tile[256];
    const int n = blockIdx.x * blockDim.x + threadIdx.x;   // 0..NSZ-1 exactly

    const float p0 = mkpts0[3 * n + 0];
    const float p1 = mkpts0[3 * n + 1];
    const float p2 = mkpts0[3 * n + 2];

    int  best = 0x7FFFFFFF;
    bool anyv = false;

    for (int t0 = 0; t0 < MSZ; t0 += 256) {
        int m = t0 + threadIdx.x;
        float vb = (float)(m >> 12);                 // m / L
        float vx = samples0[2 * m + 0];
        float vy = samples0[2 * m + 1];
        bool  v  = (jmin[m] < LSZ);
        anyv |= v;
        __syncthreads();                             // protect prior tile reads
        tile[threadIdx.x] = make_float4(vb, vx, vy, v ? 1.0f : 0.0f);
        __syncthreads();
        if (best == 0x7FFFFFFF) {                    // first match = min m
            for (int k = 0; k < 256; ++k) {
                float4 g = tile[k];
                if (g.w != 0.0f && p0 == g.x && p1 == g.y && p2 == g.z) {
                    best = t0 + k;
                    break;
                }
            }
        }
    }

    float per = 0.0f;
    bool  sel = false;
    if (best != 0x7FFFFFFF) {
        int b = best >> 12;
        int j = jmin[best];                          // < L (row was valid)
        float gx = samples1[2 * (b * LSZ + j) + 0];
        float gy = samples1[2 * (b * LSZ + j) + 1];
        float dx = gx - mkpts1[2 * n + 0];
        float dy = gy - mkpts1[2 * n + 1];
        per = sqrtf(dx * dx + dy * dy);
        sel = (per < 10.0f);
    }
    if (sel)  { atomicAdd(sel_sum, (double)per); atomicAdd(sel_cnt, 1u); }
    if (anyv) { atomicOr(any_valid, 1u); }
}

// ---------------------------------------------------------------------------
// K3: finalize -> out = [losses, coarse, fine]
// ---------------------------------------------------------------------------
__global__ void mc_finalize_kernel(const double* __restrict__ acc_d,
                                   const unsigned int* __restrict__ acc_u,
                                   float* __restrict__ out) {
    if (blockIdx.x == 0 && threadIdx.x == 0) {
        float coarse = (float)(acc_d[0] / (double)BSZ);
        unsigned int cnt  = acc_u[0];
        unsigned int anyv = acc_u[1];
        float fine = (cnt > 0u) ? (float)(acc_d[1] / (double)cnt) : 10.0f;
        if (!anyv) fine = 1e-6f;
        float losses = 1.0f * coarse + 1000.0f * fine;
        out[0] = losses;
        out[1] = coarse;
        out[2] = fine;
    }
}

// ---------------------------------------------------------------------------
// Host-side launcher
// ---------------------------------------------------------------------------
extern "C" void kernel_launch(void* const* d_in, const int* in_sizes, int n_in,
                              void* d_out, int out_size, void* d_ws, size_t ws_size,
                              hipStream_t stream) {
    const float* cm  = (const float*)d_in[0];   // [B,L,L]
    const float* gt  = (const float*)d_in[1];   // [B,L,L]
    const float* s0  = (const float*)d_in[2];   // [B,L,2]
    const float* s1  = (const float*)d_in[3];   // [B,L,2]
    const float* mk0 = (const float*)d_in[4];   // [N,3]
    const float* mk1 = (const float*)d_in[5];   // [N,2]
    (void)in_sizes; (void)n_in; (void)out_size; (void)ws_size;

    char* ws = (char*)d_ws;
    double*       acc_d = (double*)ws;              // [0]=coarse, [1]=sel_sum
    unsigned int* acc_u = (unsigned int*)(ws + 16); // [0]=cnt, [1]=any_valid
    int*          jmin  = (int*)(ws + 64);          // MSZ ints (64 KB)

    mc_init_kernel<<<(MSZ + 255) / 256, 256, 0, stream>>>(acc_d, acc_u, jmin);
    mc_coarse_kernel<<<4096, 256, 0, stream>>>((const float4*)cm, (const float4*)gt,
                                               acc_d, jmin);
    mc_fine_kernel<<<NSZ / 256, 256, 0, stream>>>(s0, s1, mk0, mk1, jmin,
                                                  acc_d + 1, acc_u, acc_u + 1);
    mc_finalize_kernel<<<1, 32, 0, stream>>>(acc_d, acc_u, (float*)d_out);
}